// JointNetwork_11003706212671
// MI455X (gfx1250) — compile-verified
//
#include <hip/hip_runtime.h>

// Problem constants (match the reference).
#define Bn 8
#define Tn 256
#define Un 64
#define Hn 512   // K dimension
#define Fn 1024  // N dimension

typedef __attribute__((ext_vector_type(2))) float v2f;
typedef __attribute__((ext_vector_type(4))) float v4f;
typedef __attribute__((ext_vector_type(8))) float v8f;

// ---------------------------------------------------------------------------
// Phase 1: C[M,N] = X[M,K] * Wp[K,N] (+ bias broadcast over rows, optional)
// f32 WMMA 16x16x4. One 16x16 output tile per wave, 8 waves per block.
//
// Fragment layouts per CDNA5 ISA (CDNA5 ISA, wave32):
//   A 16x4 : lane l in [0,16): row M=l,   VGPR0=K0, VGPR1=K1
//            lane l in [16,32): row M=l-16, VGPR0=K2, VGPR1=K3
//     -> per lane: 2 contiguous f32 at A[row][k0 + 2*(lane>>4)]   (b64 load)
//   B 4x16 : mirrored K-per-half mapping; lane supplies column N = lane&15,
//            rows k0 + 2*(lane>>4) and +1 (two N-strided f32 loads)
//   C/D    : VGPR j: lanes 0-15 -> M=j, N=lane; lanes 16-31 -> M=j+8, N=lane-16
// ---------------------------------------------------------------------------
__global__ __launch_bounds__(256) void
wmma_gemm_f32(const float* __restrict__ X, const float* __restrict__ Wp,
              const float* __restrict__ bias, float* __restrict__ C,
              int M, int addBias) {
    const int K = Hn;
    const int N = Fn;
    const int tilesN = N / 16;

    const int wave = threadIdx.x >> 5;
    const int lane = threadIdx.x & 31;
    const int half = lane >> 4;   // 0: lanes 0-15, 1: lanes 16-31
    const int l15  = lane & 15;

    const int tile  = blockIdx.x * 8 + wave;
    const int tileM = tile / tilesN;
    const int tileN = tile % tilesN;
    if (tileM * 16 >= M) return;

    // Per-lane base pointers.
    const float* __restrict__ Arow = X + (size_t)(tileM * 16 + l15) * K + 2 * half;
    const float* __restrict__ Bcol = Wp + (size_t)(2 * half) * N + (tileN * 16 + l15);

    // Accumulator: zero, or bias[n] splat across all 8 rows this lane owns
    // (bias depends only on N = tileN*16 + (lane&15)).
    v8f c;
    float binit = addBias ? bias[tileN * 16 + l15] : 0.0f;
#pragma unroll
    for (int j = 0; j < 8; ++j) c[j] = binit;

#pragma unroll 4
    for (int k0 = 0; k0 < K; k0 += 4) {
        // A fragment: 2 contiguous floats -> one global_load_b64
        v2f afrag = *(const v2f*)(Arow + k0);
        // B fragment: 2 floats strided by N
        v2f bfrag;
        bfrag.x = Bcol[(size_t)k0 * N];
        bfrag.y = Bcol[(size_t)k0 * N + N];
        // D = A*B + C   (8 args: neg_a, A, neg_b, B, c_mod, C, reuse_a, reuse_b)
        c = __builtin_amdgcn_wmma_f32_16x16x4_f32(
                false, afrag, false, bfrag, (short)0, c, false, false);
    }

    // Store the 16x16 tile per C/D layout.
    float* __restrict__ Crow =
        C + (size_t)(tileM * 16 + 8 * half) * N + (tileN * 16 + l15);
#pragma unroll
    for (int j = 0; j < 8; ++j) Crow[(size_t)j * N] = c[j];
}

// ---------------------------------------------------------------------------
// Phase 2: out[b,t,u,:] = a[b,t,:] + l[b,u,:]   (bias already folded into l)
// One block per (b,t); each thread owns one float4 column slot, holds the
// audio value in a register, and streams all U=64 rows with non-temporal
// b128 stores (write-once 512 MB must not displace a/l from L2).
//   - a row read once per block (was 64x)
//   - l rows for a fixed b are 256 KB -> L2-resident
//   - 2048 blocks x 64 NT stores/thread -> long clause-able store streams
// ---------------------------------------------------------------------------
__global__ __launch_bounds__(256) void
bcast_add(const float* __restrict__ a, const float* __restrict__ l,
          float* __restrict__ out) {
    const int bt = blockIdx.x;        // b*T + t
    const int b  = bt / Tn;
    const int i  = threadIdx.x;       // 0..255: float4 slot within the F dim

    const v4f av = ((const v4f*)(a + (size_t)bt * Fn))[i];
    const v4f* __restrict__ lp = (const v4f*)(l + (size_t)b * Un * Fn);
    v4f* __restrict__ op       = (v4f*)(out + (size_t)bt * Un * Fn);

#pragma unroll 8
    for (int u = 0; u < Un; ++u) {
        v4f lv = lp[(size_t)u * (Fn / 4) + i];
        v4f r  = av + lv;
        __builtin_nontemporal_store(r, op + (size_t)u * (Fn / 4) + i);
    }
}

extern "C" void kernel_launch(void* const* d_in, const int* in_sizes, int n_in,
                              void* d_out, int out_size, void* d_ws, size_t ws_size,
                              hipStream_t stream) {
    (void)in_sizes; (void)n_in; (void)out_size; (void)ws_size;

    const float* audio = (const float*)d_in[0];   // (B,T,H)
    const float* label = (const float*)d_in[1];   // (B,U,H)
    const float* W     = (const float*)d_in[2];   // (2H,F): Wa=W[0:H], Wl=W[H:2H]
    const float* bias  = (const float*)d_in[3];   // (F,)

    float* ws_a = (float*)d_ws;                       // (B*T, F) = 2048x1024
    float* ws_l = ws_a + (size_t)(Bn * Tn) * Fn;      // (B*U, F) =  512x1024

    // audio GEMM: M=2048 -> 128x64 = 8192 tiles -> 1024 blocks of 8 waves
    wmma_gemm_f32<<<(Bn * Tn / 16) * (Fn / 16) / 8, 256, 0, stream>>>(
        audio, W, nullptr, ws_a, Bn * Tn, 0);
    // label GEMM (+bias): M=512 -> 32x64 = 2048 tiles -> 256 blocks
    wmma_gemm_f32<<<(Bn * Un / 16) * (Fn / 16) / 8, 256, 0, stream>>>(
        label, W + (size_t)Hn * Fn, bias, ws_l, Bn * Un, 1);

    // broadcast add: one block per (b,t), loops over all u
    bcast_add<<<Bn * Tn, 256, 0, stream>>>(ws_a, ws_l, (float*)d_out);
}